// BessKGE_24197845745912
// MI455X (gfx1250) — compile-verified
//
#include <hip/hip_runtime.h>

typedef __attribute__((ext_vector_type(2))) float v2f;
typedef __attribute__((ext_vector_type(8))) float v8f;

#define N_SHARD 4
#define PPP     512
#define NEG_N   1024
#define EMB     256
#define M_TOT   (N_SHARD * PPP)   // 2048 rows of q / positive scores

// Device-pass-only availability check (host pass has no amdgcn builtins).
#if defined(__HIP_DEVICE_COMPILE__)
#if !__has_builtin(__builtin_amdgcn_wmma_f32_16x16x4_f32)
#error "DEVICE: wmma_f32_16x16x4_f32 builtin not available on this toolchain"
#endif
#endif

// ---------------------------------------------------------------------------
// Kernel 1: q[m][e] = ent[head[m]][e] * rel[relation[m]][e]   (to workspace)
//           pos[m]  = sum_e q[m][e] * ent[tail[m]][e]         (to d_out[0:2048])
// One block per row m, 256 threads = one thread per embedding element.
// ---------------------------------------------------------------------------
__global__ __launch_bounds__(EMB) void kge_prep(
    const int* __restrict__ head, const int* __restrict__ relation,
    const int* __restrict__ tail,
    const float* __restrict__ ent, const float* __restrict__ rel,
    float* __restrict__ q, float* __restrict__ pos)
{
    const int m = blockIdx.x;
    const int e = threadIdx.x;
    const int h = head[m];
    const int r = relation[m];
    const int t = tail[m];
    const float hv = ent[(size_t)h * EMB + e];
    const float rv = rel[(size_t)r * EMB + e];
    const float tv = ent[(size_t)t * EMB + e];
    const float qv = hv * rv;
    q[(size_t)m * EMB + e] = qv;

    __shared__ float red[EMB];
    red[e] = qv * tv;
    __syncthreads();
    #pragma unroll
    for (int s = EMB / 2; s > 0; s >>= 1) {
        if (e < s) red[e] += red[e + s];
        __syncthreads();
    }
    if (e == 0) pos[m] = red[0];
}

// ---------------------------------------------------------------------------
// Kernel 2: gather negative embeddings (only negative[0,0,:] is used by the
// reference) into a dense 1024x256 buffer so GEMM B-loads are contiguous.
// ---------------------------------------------------------------------------
__global__ __launch_bounds__(EMB) void kge_gather_neg(
    const int* __restrict__ negative, const float* __restrict__ ent,
    float* __restrict__ negbuf)
{
    const int c = blockIdx.x;      // 0..1023
    const int e = threadIdx.x;     // 0..255
    const int idx = negative[c];   // negative[0,0,c] (flat layout (4,1,1024))
    negbuf[(size_t)c * EMB + e] = ent[(size_t)idx * EMB + e];
}

// ---------------------------------------------------------------------------
// Kernel 3: S(2048x1024) = q(2048x256) * negbuf(1024x256)^T via
// v_wmma_f32_16x16x4_f32. Wave computes a 16x32 tile (2 accumulators share
// one A fragment). Block = 8 waves -> 64x64 tile. Output is written 4x
// column-tiled (the reference's broadcast over shards) with nontemporal
// stores: out_neg[m, i*1024 + c] = S[m, c] for i = 0..3.
// ---------------------------------------------------------------------------
__global__ __launch_bounds__(256) void kge_gemm(
    const float* __restrict__ q, const float* __restrict__ negbuf,
    float* __restrict__ out_neg)
{
    const int lane = threadIdx.x & 31;
    const int wave = threadIdx.x >> 5;
    const int half = lane >> 4;     // 0: K 0..1 / rows M0..7 ; 1: K 2..3 / rows M8..15
    const int lrow = lane & 15;

    const int m0 = blockIdx.x * 64 + (wave >> 1) * 16;  // 2048 / 64 = 32 blocks in x
    const int n0 = blockIdx.y * 64 + (wave & 1) * 32;   // 1024 / 64 = 16 blocks in y

    // A fragment: lane holds q[m0+lrow][k + 2*half .. +1]  (8-byte aligned)
    const float* aptr  = q      + (size_t)(m0 + lrow)      * EMB + 2 * half;
    // B fragments: lane holds negbuf[n0(+16)+lrow][k + 2*half .. +1]
    const float* bptr0 = negbuf + (size_t)(n0 + lrow)      * EMB + 2 * half;
    const float* bptr1 = negbuf + (size_t)(n0 + 16 + lrow) * EMB + 2 * half;

    v8f acc0 = {};
    v8f acc1 = {};

    #pragma unroll 8
    for (int k = 0; k < EMB; k += 4) {
        v2f a  = *(const v2f*)(aptr  + k);
        v2f b0 = *(const v2f*)(bptr0 + k);
        v2f b1 = *(const v2f*)(bptr1 + k);
        acc0 = __builtin_amdgcn_wmma_f32_16x16x4_f32(
            /*neg_a=*/false, a, /*neg_b=*/false, b0,
            /*c_mod=*/(short)0, acc0, /*reuse_a=*/false, /*reuse_b=*/false);
        acc1 = __builtin_amdgcn_wmma_f32_16x16x4_f32(
            /*neg_a=*/false, a, /*neg_b=*/false, b1,
            /*c_mod=*/(short)0, acc1, /*reuse_a=*/false, /*reuse_b=*/false);
    }

    // C/D layout: VGPR v holds (M = m0 + v + 8*half, N = n0 + (lane&15)).
    #pragma unroll
    for (int v = 0; v < 8; ++v) {
        const int row = m0 + v + 8 * half;
        float* rbase = out_neg + (size_t)row * (N_SHARD * NEG_N);
        #pragma unroll
        for (int i = 0; i < N_SHARD; ++i) {
            __builtin_nontemporal_store(acc0[v], rbase + i * NEG_N + n0 + lrow);
            __builtin_nontemporal_store(acc1[v], rbase + i * NEG_N + n0 + 16 + lrow);
        }
    }
}

// ---------------------------------------------------------------------------
extern "C" void kernel_launch(void* const* d_in, const int* in_sizes, int n_in,
                              void* d_out, int out_size, void* d_ws, size_t ws_size,
                              hipStream_t stream)
{
    const int*   head     = (const int*)d_in[0];
    const int*   relation = (const int*)d_in[1];
    const int*   tail     = (const int*)d_in[2];
    const int*   negative = (const int*)d_in[3];
    const float* ent      = (const float*)d_in[4];
    const float* rel      = (const float*)d_in[5];

    float* out     = (float*)d_out;
    float* pos     = out;             // 2048 positive scores
    float* out_neg = out + M_TOT;     // 2048 x 4096 negative scores

    float* q      = (float*)d_ws;                 // 2048 x 256 f32 (2 MB)
    float* negbuf = q + (size_t)M_TOT * EMB;      // 1024 x 256 f32 (1 MB)

    hipLaunchKernelGGL(kge_prep, dim3(M_TOT), dim3(EMB), 0, stream,
                       head, relation, tail, ent, rel, q, pos);
    hipLaunchKernelGGL(kge_gather_neg, dim3(NEG_N), dim3(EMB), 0, stream,
                       negative, ent, negbuf);
    hipLaunchKernelGGL(kge_gemm, dim3(M_TOT / 64, NEG_N / 64), dim3(256), 0, stream,
                       q, negbuf, out_neg);
}